// FMLayer_77558519431755
// MI455X (gfx1250) — compile-verified
//
#include <hip/hip_runtime.h>
#include <stdint.h>

#define EMBED 64
#define CHUNK 128   // entries per staged chunk (double-buffered)

// ---- CDNA5 async global->LDS (ASYNCcnt path), per CDNA5 ISA ----
// GVS mode: 64-bit SGPR base plus 32-bit per-lane VGPR byte offset; data goes to the lane's LDS address
__device__ __forceinline__ uint32_t lds_off(const void* p) {
    // flat LDS addresses keep the aperture in bits [63:32]; low 32 bits are the
    // workgroup-relative LDS byte address (hardware adds LDS_BASE).
    return (uint32_t)(uintptr_t)p;
}
__device__ __forceinline__ void async_ld_b128(uint32_t lds, uint64_t sbase, uint32_t voff) {
    asm volatile("global_load_async_to_lds_b128 %0, %1, %2"
                 :: "v"(lds), "v"(voff), "s"(sbase) : "memory");
}
__device__ __forceinline__ void wait_async0() {
    asm volatile("s_wait_asynccnt 0x0" ::: "memory");
}

// One 64-thread block (2 wave32) per batch row; batch_ids sorted -> contiguous run.
// Wave w consumes entries i ≡ w (mod 2); lane l owns embed dims {2l, 2l+1} via
// float2 (b64) gathers: 32 lanes x 8B = one full 256B embedding row per entry.
__global__ __launch_bounds__(64) void fm_layer_rows(
    const float* __restrict__ emb,   // [VOCAB, EMBED]
    const float* __restrict__ vals,  // [NNZ]
    const int*   __restrict__ bids,  // [NNZ], sorted
    const int*   __restrict__ fids,  // [NNZ]
    float*       __restrict__ out,   // [BATCH, EMBED]
    int nnz)
{
    const int b = blockIdx.x;
    const int t = threadIdx.x;
    const int w = t >> 5;   // wave 0/1
    const int l = t & 31;   // lane in wave

    // block-uniform binary searches over sorted batch_ids (scalar s_load path)
    int lo = 0, hi = nnz;
    while (lo < hi) { int mid = (lo + hi) >> 1; if (bids[mid] <  b) lo = mid + 1; else hi = mid; }
    const int start = lo;
    lo = start; hi = nnz;
    while (lo < hi) { int mid = (lo + hi) >> 1; if (bids[mid] <= b) lo = mid + 1; else hi = mid; }
    const int end = lo;

    // 16B alignment required: async b128 writes 16B/lane into these.
    __shared__ __align__(16) float s_v[2][CHUNK];
    __shared__ __align__(16) int   s_f[2][CHUNK];
    __shared__ __align__(16) float pS[2][EMBED];
    __shared__ __align__(16) float pQ[2][EMBED];

    float accS0 = 0.f, accS1 = 0.f, accQ0 = 0.f, accQ1 = 0.f;

    const uint64_t vals_base = (uint64_t)(uintptr_t)vals;
    const uint64_t fids_base = (uint64_t)(uintptr_t)fids;

    // Stage one CHUNK: wave0 lanes DMA 128 vals (b128 = 4 floats/lane),
    // wave1 lanes DMA 128 feat_ids. Addresses clamped inside the arrays;
    // LDS slots >= count are never consumed.
    auto stage = [&](int buf, int base) {
        int e  = base + 4 * l;
        int ec = (e <= nnz - 4) ? e : (nnz - 4);
        uint32_t voff = (uint32_t)(ec * 4);
        if (w == 0) async_ld_b128(lds_off(&s_v[buf][4 * l]), vals_base, voff);
        else        async_ld_b128(lds_off(&s_f[buf][4 * l]), fids_base, voff);
    };

    // Warm L2 for chunk at nbase while the previous chunk is being consumed.
    auto prefetch_rows = [&](int nbase) {
        int i0 = nbase + t;       int c0 = (i0 < end) ? i0 : end - 1;
        int i1 = nbase + 64 + t;  int c1 = (i1 < end) ? i1 : end - 1;
        int f0 = fids[c0], f1 = fids[c1];
        const float* r0 = emb + (size_t)f0 * EMBED;
        const float* r1 = emb + (size_t)f1 * EMBED;
        __builtin_prefetch(r0,      0, 1);  __builtin_prefetch(r0 + 32, 0, 1);
        __builtin_prefetch(r1,      0, 1);  __builtin_prefetch(r1 + 32, 0, 1);
    };

    auto consume = [&](int buf, int count) {
        int i = w;
        for (; i + 6 < count; i += 8) {   // 4 entries per wave in flight
            float v0 = s_v[buf][i],     v1 = s_v[buf][i + 2];
            float v2 = s_v[buf][i + 4], v3 = s_v[buf][i + 6];
            int   f0 = s_f[buf][i],     f1 = s_f[buf][i + 2];
            int   f2 = s_f[buf][i + 4], f3 = s_f[buf][i + 6];
            float2 x0 = *(const float2*)(emb + (size_t)f0 * EMBED + 2 * l);
            float2 x1 = *(const float2*)(emb + (size_t)f1 * EMBED + 2 * l);
            float2 x2 = *(const float2*)(emb + (size_t)f2 * EMBED + 2 * l);
            float2 x3 = *(const float2*)(emb + (size_t)f3 * EMBED + 2 * l);
            float a0x = v0 * x0.x, a0y = v0 * x0.y;
            float a1x = v1 * x1.x, a1y = v1 * x1.y;
            float a2x = v2 * x2.x, a2y = v2 * x2.y;
            float a3x = v3 * x3.x, a3y = v3 * x3.y;
            accS0 += a0x; accS1 += a0y;
            accS0 += a1x; accS1 += a1y;
            accS0 += a2x; accS1 += a2y;
            accS0 += a3x; accS1 += a3y;
            accQ0 = fmaf(a0x, a0x, accQ0); accQ1 = fmaf(a0y, a0y, accQ1);
            accQ0 = fmaf(a1x, a1x, accQ0); accQ1 = fmaf(a1y, a1y, accQ1);
            accQ0 = fmaf(a2x, a2x, accQ0); accQ1 = fmaf(a2y, a2y, accQ1);
            accQ0 = fmaf(a3x, a3x, accQ0); accQ1 = fmaf(a3y, a3y, accQ1);
        }
        for (; i < count; i += 2) {
            float v = s_v[buf][i];
            int   f = s_f[buf][i];
            float2 x = *(const float2*)(emb + (size_t)f * EMBED + 2 * l);
            float ax = v * x.x, ay = v * x.y;
            accS0 += ax; accS1 += ay;
            accQ0 = fmaf(ax, ax, accQ0);
            accQ1 = fmaf(ay, ay, accQ1);
        }
    };

    if (start < end) {
        stage(0, start);
        prefetch_rows(start);
        wait_async0();
        __syncthreads();
        int buf = 0;
        for (int base = start; base < end; base += CHUNK) {
            int nbase = base + CHUNK;
            if (nbase < end) {            // block-uniform
                stage(buf ^ 1, nbase);    // DMA next chunk while consuming this one
                prefetch_rows(nbase);     // and warm L2 for its embedding rows
            }
            consume(buf, min(CHUNK, end - base));
            wait_async0();
            __syncthreads();
            buf ^= 1;
        }
    }

    // combine the two waves' partials (wave w holds entries i%2==w)
    pS[w][2 * l] = accS0;  pS[w][2 * l + 1] = accS1;
    pQ[w][2 * l] = accQ0;  pQ[w][2 * l + 1] = accQ1;
    __syncthreads();
    float S = pS[0][t] + pS[1][t];
    float Q = pQ[0][t] + pQ[1][t];
    out[(size_t)b * EMBED + t] = 0.5f * (S * S - Q);
}

extern "C" void kernel_launch(void* const* d_in, const int* in_sizes, int n_in,
                              void* d_out, int out_size, void* d_ws, size_t ws_size,
                              hipStream_t stream) {
    const float* emb  = (const float*)d_in[0];
    const float* vals = (const float*)d_in[1];
    const int*   bids = (const int*)  d_in[2];
    const int*   fids = (const int*)  d_in[3];
    float*       out  = (float*)d_out;

    const int nnz   = in_sizes[1];
    const int batch = out_size / EMBED;   // 4096

    fm_layer_rows<<<batch, 64, 0, stream>>>(emb, vals, bids, fids, out, nnz);
}